// GNNModel_2680059593455
// MI455X (gfx1250) — compile-verified
//
#include <hip/hip_runtime.h>
#include <math.h>

typedef __attribute__((ext_vector_type(2))) float v2f;
typedef __attribute__((ext_vector_type(8))) float v8f;

// Fixed problem geometry (reference): N=100000 (divisible by 32), D_IN=D_HID=64, D_OUT=34.

// ---------------- degree / normalization ----------------
__global__ void k_deg_init(float* __restrict__ deg, int n) {
    int i = blockIdx.x * blockDim.x + threadIdx.x;
    if (i < n) deg[i] = 1.0f;               // self-loop contributes 1
}

__global__ void k_deg_count(const int* __restrict__ dst, float* __restrict__ deg, int E) {
    int e = blockIdx.x * blockDim.x + threadIdx.x;
    if (e < E) atomicAdd(&deg[dst[e]], 1.0f);
}

__global__ void k_dinv(const float* __restrict__ deg, float* __restrict__ dinv, int n) {
    int i = blockIdx.x * blockDim.x + threadIdx.x;
    if (i < n) dinv[i] = rsqrtf(deg[i]);    // deg >= 1 always (self-loops)
}

// ---------------- GEMM1: xw[N x 64] = x[N x 64] @ W1[64 x 64] (fp32 WMMA) ----------------
// 8 waves/block; wave w -> M-tile (w>>2), N-tile (w&3). Block covers 32 rows x 64 cols.
// A 16x4 f32 layout: lanes 0-15 hold row M=lane, K={kb,kb+1}; lanes 16-31 hold K={kb+2,kb+3}.
// B 4x16 f32 layout mirrored: lane holds column N, two K values per VGPR pair.
__global__ void __launch_bounds__(256) k_gemm64(const float* __restrict__ A,
                                                const float* __restrict__ B,
                                                float* __restrict__ C) {
    const int wave = threadIdx.x >> 5;
    const int lane = threadIdx.x & 31;
    const int half = lane >> 4;             // 0: K+0/1, 1: K+2/3
    const int lm   = lane & 15;
    const int m0   = blockIdx.x * 32 + (wave >> 2) * 16;
    const int n0   = (wave & 3) * 16;
    const float* arow = A + (size_t)(m0 + lm) * 64;
    const int bcol = n0 + lm;

    v8f acc = {0.f,0.f,0.f,0.f,0.f,0.f,0.f,0.f};
#pragma unroll
    for (int kk = 0; kk < 16; ++kk) {
        const int kb = kk * 4 + half * 2;   // even -> 8B aligned
        v2f a = *(const v2f*)(arow + kb);
        v2f b;
        b.x = B[(size_t)kb * 64 + bcol];
        b.y = B[(size_t)(kb + 1) * 64 + bcol];
        acc = __builtin_amdgcn_wmma_f32_16x16x4_f32(false, a, false, b,
                                                    (short)0, acc, false, false);
    }
    // D layout: VGPR i -> (M = i + 8*half, N = lm)
#pragma unroll
    for (int i = 0; i < 8; ++i)
        C[(size_t)(m0 + half * 8 + i) * 64 + n0 + lm] = acc[i];
}

// ---------------- layer-1 self-loop + bias, then edge scatter ----------------
__global__ void k_init_h(const float* __restrict__ xw, const float* __restrict__ dinv,
                         const float* __restrict__ b1, float* __restrict__ h, int n) {
    int idx = blockIdx.x * blockDim.x + threadIdx.x;
    if (idx >= n * 64) return;
    int i = idx >> 6, c = idx & 63;
    float di = dinv[i];
    h[idx] = b1[c] + di * di * xw[idx];
}

__global__ void k_scatter64(const int* __restrict__ src, const int* __restrict__ dst,
                            const float* __restrict__ dinv, const float* __restrict__ xw,
                            float* __restrict__ h, int E) {
    int idx = blockIdx.x * blockDim.x + threadIdx.x;
    if (idx >= E * 16) return;              // 4 channels per thread
    int e = idx >> 4;
    int c = (idx & 15) << 2;
    int s = src[e], d = dst[e];
    float w = dinv[s] * dinv[d];
    const float4 v = *(const float4*)(xw + (size_t)s * 64 + c);
    float* o = h + (size_t)d * 64 + c;
    atomicAdd(o + 0, w * v.x);
    atomicAdd(o + 1, w * v.y);
    atomicAdd(o + 2, w * v.z);
    atomicAdd(o + 3, w * v.w);
}

// ---------------- fused relu + softmax over 64 channels (one wave per node) ----------------
__global__ void __launch_bounds__(128) k_relu_softmax(float* __restrict__ h, int n) {
    int node = blockIdx.x * 4 + (threadIdx.x >> 5);
    int lane = threadIdx.x & 31;
    if (node >= n) return;                  // wave-uniform
    float* row = h + (size_t)node * 64;
    float v0 = fmaxf(row[lane], 0.0f);
    float v1 = fmaxf(row[lane + 32], 0.0f);
    float m = fmaxf(v0, v1);
#pragma unroll
    for (int off = 16; off > 0; off >>= 1) m = fmaxf(m, __shfl_xor(m, off, 32));
    float e0 = __expf(v0 - m), e1 = __expf(v1 - m);
    float s = e0 + e1;
#pragma unroll
    for (int off = 16; off > 0; off >>= 1) s += __shfl_xor(s, off, 32);
    float inv = 1.0f / s;
    row[lane]      = e0 * inv;
    row[lane + 32] = e1 * inv;
}

// ---------------- GEMM2: hw2[N x 34] = h[N x 64] @ W2[64 x 34] (N padded to 48) ----------------
// 6 waves/block: 2 M-tiles x 3 N-tiles; B loads zero-padded for col >= 34.
__global__ void __launch_bounds__(192) k_gemm34(const float* __restrict__ A,
                                                const float* __restrict__ B,
                                                float* __restrict__ C) {
    const int wave = threadIdx.x >> 5;      // 0..5
    const int lane = threadIdx.x & 31;
    const int half = lane >> 4;
    const int lm   = lane & 15;
    const int m0   = blockIdx.x * 32 + (wave / 3) * 16;
    const int n0   = (wave % 3) * 16;
    const int col  = n0 + lm;
    const bool ok  = (col < 34);
    const float* arow = A + (size_t)(m0 + lm) * 64;

    v8f acc = {0.f,0.f,0.f,0.f,0.f,0.f,0.f,0.f};
#pragma unroll
    for (int kk = 0; kk < 16; ++kk) {
        const int kb = kk * 4 + half * 2;
        v2f a = *(const v2f*)(arow + kb);
        v2f b;
        b.x = ok ? B[(size_t)kb * 34 + col]       : 0.0f;
        b.y = ok ? B[(size_t)(kb + 1) * 34 + col] : 0.0f;
        acc = __builtin_amdgcn_wmma_f32_16x16x4_f32(false, a, false, b,
                                                    (short)0, acc, false, false);
    }
    if (ok) {
#pragma unroll
        for (int i = 0; i < 8; ++i)
            C[(size_t)(m0 + half * 8 + i) * 34 + col] = acc[i];
    }
}

// ---------------- layer-2 self-loop + bias, then edge scatter ----------------
__global__ void k_init_out(const float* __restrict__ hw2, const float* __restrict__ dinv,
                           const float* __restrict__ b2, float* __restrict__ out, int n) {
    int idx = blockIdx.x * blockDim.x + threadIdx.x;
    if (idx >= n * 34) return;
    int i = idx / 34, c = idx - i * 34;
    float di = dinv[i];
    out[idx] = b2[c] + di * di * hw2[idx];
}

__global__ void k_scatter34(const int* __restrict__ src, const int* __restrict__ dst,
                            const float* __restrict__ dinv, const float* __restrict__ hw2,
                            float* __restrict__ out, int E) {
    int idx = blockIdx.x * blockDim.x + threadIdx.x;
    if (idx >= E * 34) return;
    int e = idx / 34, c = idx - e * 34;
    int s = src[e], d = dst[e];
    float w = dinv[s] * dinv[d];
    atomicAdd(&out[(size_t)d * 34 + c], w * hw2[(size_t)s * 34 + c]);
}

extern "C" void kernel_launch(void* const* d_in, const int* in_sizes, int n_in,
                              void* d_out, int out_size, void* d_ws, size_t ws_size,
                              hipStream_t stream) {
    const float* x  = (const float*)d_in[0];
    const int*   ei = (const int*)  d_in[1];
    const float* W1 = (const float*)d_in[2];
    const float* b1 = (const float*)d_in[3];
    const float* W2 = (const float*)d_in[4];
    const float* b2 = (const float*)d_in[5];
    float* out = (float*)d_out;

    const int N = in_sizes[0] / 64;     // 100000 (divisible by 32)
    const int E = in_sizes[1] / 2;      // 1200000
    const int* src = ei;
    const int* dst = ei + E;

    // workspace layout (floats): deg[N] dinv[N] xw[N*64] h[N*64] hw2[N*34]  (~66 MB)
    float* deg  = (float*)d_ws;
    float* dinv = deg  + N;
    float* xw   = dinv + N;
    float* h    = xw   + (size_t)N * 64;
    float* hw2  = h    + (size_t)N * 64;

    const int T = 256;
    // 1) symmetric normalization
    k_deg_init <<<(N + T - 1) / T, T, 0, stream>>>(deg, N);
    k_deg_count<<<(E + T - 1) / T, T, 0, stream>>>(dst, deg, E);
    k_dinv     <<<(N + T - 1) / T, T, 0, stream>>>(deg, dinv, N);
    // 2) xw = x @ W1   (fp32 WMMA)
    k_gemm64<<<N / 32, 256, 0, stream>>>(x, W1, xw);
    // 3) h = b1 + dinv^2 * xw  (self-loop), then edge scatter
    k_init_h   <<<((size_t)N * 64 + T - 1) / T, T, 0, stream>>>(xw, dinv, b1, h, N);
    k_scatter64<<<((size_t)E * 16 + T - 1) / T, T, 0, stream>>>(src, dst, dinv, xw, h, E);
    // 4) relu + softmax (rowwise)
    k_relu_softmax<<<(N + 3) / 4, 128, 0, stream>>>(h, N);
    // 5) hw2 = h @ W2  (fp32 WMMA, N padded 34->48)
    k_gemm34<<<N / 32, 192, 0, stream>>>(h, W2, hw2);
    // 6) out = b2 + dinv^2 * hw2, then edge scatter
    k_init_out <<<((size_t)N * 34 + T - 1) / T, T, 0, stream>>>(hw2, dinv, b2, out, N);
    k_scatter34<<<((size_t)E * 34 + T - 1) / T, T, 0, stream>>>(src, dst, dinv, hw2, out, E);
}